// SpectralAttention_71906342470265
// MI455X (gfx1250) — compile-verified
//
#include <hip/hip_runtime.h>

// Spectral band attention, algebraically collapsed.
//
// tokens[n,b,:] = x[n,b]*w_embed + (b_embed + band_pos[b])  is affine in the
// scalar x[n,b], so Q/K/V are affine in it too, and the whole attention
// collapses to tiny precomputed tensors:
//   logit_h[b,b'] = scale*(x_b x_b' A_h + x_b U_h[b'] + x_b' V_h[b] + C_h[b,b'])
//   delta[b]      = sum_h sum_b' softmax_row * (x_b' s_h + t_h[b']) + bo
// Main pass: 96 B HBM traffic + ~2 KFLOP per token -> pure bandwidth kernel
// (6.3 MB @ 23.3 TB/s ~ 0.3 us). The only real matrix math is the precompute
// R_p = T(16x128) @ Wp^T(128x128), done with V_WMMA_F32_16X16X4_F32 in f32.

typedef __attribute__((ext_vector_type(2))) float v2f;
typedef __attribute__((ext_vector_type(8))) float v8f;

#define NBANDS 12
#define DMODEL 128
#define HDIM   64

// ---------------------------------------------------------------------------
// Precompute kernel: one block, 8 waves. Wave w computes the 16-column tile
// d0=w*16 of R_p = T @ Wp^T for p in {Q,K,V} using f32 WMMA, then all threads
// reduce the 64-wide head dot-products into the 365-float constant table.
// ---------------------------------------------------------------------------
__global__ __launch_bounds__(256) void spectral_precompute_kernel(
    const float* __restrict__ w_embed, const float* __restrict__ b_embed,
    const float* __restrict__ band_pos,
    const float* __restrict__ Wq, const float* __restrict__ bq,
    const float* __restrict__ Wk, const float* __restrict__ bk,
    const float* __restrict__ Wv, const float* __restrict__ bv,
    const float* __restrict__ Wo, const float* __restrict__ bo,
    float* __restrict__ ws)
{
    __shared__ float T[16 * DMODEL];        // rows 0..11: c[b]; row 12: w_embed
    __shared__ float R[3][16 * DMODEL];     // R[p] = T @ Wp^T

    const int tid = threadIdx.x;

    // Build T in LDS.
    for (int i = tid; i < 16 * DMODEL; i += 256) {
        int row = i >> 7, d = i & 127;
        float v = 0.0f;
        if (row < NBANDS)      v = b_embed[d] + band_pos[row * DMODEL + d];
        else if (row == 12)    v = w_embed[d];
        T[i] = v;
    }
    __syncthreads();

    const float* Wmat[3] = { Wq, Wk, Wv };
    const int wave = tid >> 5;
    const int lane = tid & 31;
    const int m    = lane & 15;               // A row / B-D column within tile
    const int ksel = (lane >= 16) ? 2 : 0;    // K pair selector for A/B frags
    const int d0   = wave * 16;               // output column tile

    for (int p = 0; p < 3; ++p) {
        const float* W = Wmat[p];
        v8f c = { 0.f, 0.f, 0.f, 0.f, 0.f, 0.f, 0.f, 0.f };
        for (int k0 = 0; k0 < DMODEL; k0 += 4) {
            v2f a, b;
            // A (16x4 f32): lanes 0-15 hold K={k0,k0+1}, lanes 16-31 K={k0+2,k0+3}
            a.x = T[m * DMODEL + k0 + ksel];
            a.y = T[m * DMODEL + k0 + ksel + 1];
            // B (4x16 f32): B[k][n] = W[(d0+n)*128 + k0+k]; row-striped like C/D
            b.x = W[(d0 + m) * DMODEL + k0 + ksel];
            b.y = W[(d0 + m) * DMODEL + k0 + ksel + 1];
            c = __builtin_amdgcn_wmma_f32_16x16x4_f32(
                    false, a, false, b, (short)0, c, false, false);
        }
        // D layout: VGPR j -> row j (lanes 0-15) / row j+8 (lanes 16-31), col m
        const int rbase = (lane >= 16) ? 8 : 0;
        for (int j = 0; j < 8; ++j)
            R[p][(j + rbase) * DMODEL + d0 + m] = c[j];
    }
    __syncthreads();

    // Derived constants (pre-scaled by 1/sqrt(64) where they feed logits):
    //   ws[0..1]     A[h]
    //   ws[2..25]    U[h][b']
    //   ws[26..49]   V[h][b]
    //   ws[50..337]  C[h][b][b']
    //   ws[338..339] s[h]
    //   ws[340..363] t[h][b']
    //   ws[364]      bo
    const float scale = 0.125f;
    for (int i = tid; i < 364; i += 256) {
        float acc = 0.0f;
        if (i < 2) {                                   // A[h] = qw_h . kw_h
            int dbase = i * HDIM;
            for (int d = 0; d < HDIM; ++d)
                acc += R[0][12 * DMODEL + dbase + d] * R[1][12 * DMODEL + dbase + d];
            ws[i] = acc * scale;
        } else if (i < 26) {                           // U[h][b'] = qw_h . kc_h[b']
            int j = i - 2, h = j / 12, bp = j % 12, dbase = h * HDIM;
            for (int d = 0; d < HDIM; ++d)
                acc += R[0][12 * DMODEL + dbase + d] *
                       (R[1][bp * DMODEL + dbase + d] + bk[dbase + d]);
            ws[i] = acc * scale;
        } else if (i < 50) {                           // V[h][b] = qc_h[b] . kw_h
            int j = i - 26, h = j / 12, b = j % 12, dbase = h * HDIM;
            for (int d = 0; d < HDIM; ++d)
                acc += (R[0][b * DMODEL + dbase + d] + bq[dbase + d]) *
                       R[1][12 * DMODEL + dbase + d];
            ws[i] = acc * scale;
        } else if (i < 338) {                          // C[h][b][b'] = qc.kc
            int j = i - 50, h = j / 144, r = j % 144, b = r / 12, bp = r % 12;
            int dbase = h * HDIM;
            for (int d = 0; d < HDIM; ++d)
                acc += (R[0][b  * DMODEL + dbase + d] + bq[dbase + d]) *
                       (R[1][bp * DMODEL + dbase + d] + bk[dbase + d]);
            ws[i] = acc * scale;
        } else if (i < 340) {                          // s[h] = vw_h . Wo_h
            int dbase = (i - 338) * HDIM;
            for (int d = 0; d < HDIM; ++d)
                acc += R[2][12 * DMODEL + dbase + d] * Wo[dbase + d];
            ws[i] = acc;
        } else {                                       // t[h][b'] = vc_h[b'] . Wo_h
            int j = i - 340, h = j / 12, bp = j % 12, dbase = h * HDIM;
            for (int d = 0; d < HDIM; ++d)
                acc += (R[2][bp * DMODEL + dbase + d] + bv[dbase + d]) * Wo[dbase + d];
            ws[i] = acc;
        }
    }
    if (tid == 0) ws[364] = bo[0];
}

// ---------------------------------------------------------------------------
// Main streaming kernel: one thread per token. 48 B in + 48 B out per token,
// constants broadcast from 1.5 KB of LDS. Pure HBM-bandwidth bound.
// ---------------------------------------------------------------------------
__global__ __launch_bounds__(256) void spectral_attn_main_kernel(
    const float* __restrict__ x, const float* __restrict__ ws,
    float* __restrict__ out, int ntok)
{
    __shared__ float cs[365];
    for (int i = threadIdx.x; i < 365; i += 256) cs[i] = ws[i];
    __syncthreads();

    const int n = blockIdx.x * blockDim.x + threadIdx.x;
    if (n >= ntok) return;

    // 12 contiguous f32 per token, 16-B aligned -> three b128 loads.
    const float4* xp = (const float4*)(x + (size_t)n * NBANDS);
    float4 x0 = xp[0], x1 = xp[1], x2 = xp[2];
    float xv[NBANDS] = { x0.x, x0.y, x0.z, x0.w,
                         x1.x, x1.y, x1.z, x1.w,
                         x2.x, x2.y, x2.z, x2.w };

    const float bo = cs[364];
    float acc[NBANDS];
#pragma unroll
    for (int b = 0; b < NBANDS; ++b) acc[b] = xv[b] + bo;

#pragma unroll
    for (int h = 0; h < 2; ++h) {
        const float  Ah = cs[h];
        const float* Uh = cs + 2   + h * 12;
        const float* Vh = cs + 26  + h * 12;
        const float* Ch = cs + 50  + h * 144;
        const float  sh = cs[338 + h];
        const float* th = cs + 340 + h * 12;

        float g[NBANDS], w[NBANDS];
#pragma unroll
        for (int bp = 0; bp < NBANDS; ++bp) {
            g[bp] = xv[bp] * Ah + Uh[bp];     // x_b * g[b'] gives A + U terms
            w[bp] = xv[bp] * sh + th[bp];     // collapsed value row
        }
#pragma unroll
        for (int b = 0; b < NBANDS; ++b) {
            const float xb = xv[b], vb = Vh[b];
            float l[NBANDS], mx = -3.0e38f;
#pragma unroll
            for (int bp = 0; bp < NBANDS; ++bp) {
                float t = xb * g[bp] + xv[bp] * vb + Ch[b * 12 + bp];
                l[bp] = t;
                mx = fmaxf(mx, t);
            }
            float sum = 0.0f, dot = 0.0f;
#pragma unroll
            for (int bp = 0; bp < NBANDS; ++bp) {
                float e = __expf(l[bp] - mx);
                sum += e;
                dot += e * w[bp];
            }
            acc[b] += dot / sum;
        }
    }

    float4* op = (float4*)(out + (size_t)n * NBANDS);
    op[0] = make_float4(acc[0], acc[1], acc[2],  acc[3]);
    op[1] = make_float4(acc[4], acc[5], acc[6],  acc[7]);
    op[2] = make_float4(acc[8], acc[9], acc[10], acc[11]);
}

extern "C" void kernel_launch(void* const* d_in, const int* in_sizes, int n_in,
                              void* d_out, int out_size, void* d_ws, size_t ws_size,
                              hipStream_t stream) {
    const float* x        = (const float*)d_in[0];
    const float* w_embed  = (const float*)d_in[1];
    const float* b_embed  = (const float*)d_in[2];
    const float* band_pos = (const float*)d_in[3];
    const float* Wq       = (const float*)d_in[4];
    const float* bq       = (const float*)d_in[5];
    const float* Wk       = (const float*)d_in[6];
    const float* bk       = (const float*)d_in[7];
    const float* Wv       = (const float*)d_in[8];
    const float* bv       = (const float*)d_in[9];
    const float* Wo       = (const float*)d_in[10];
    const float* bo       = (const float*)d_in[11];
    float* ws  = (float*)d_ws;
    float* out = (float*)d_out;

    const int ntok = in_sizes[0] / NBANDS;   // 65536 tokens

    spectral_precompute_kernel<<<1, 256, 0, stream>>>(
        w_embed, b_embed, band_pos, Wq, bq, Wk, bk, Wv, bv, Wo, bo, ws);

    spectral_attn_main_kernel<<<(ntok + 255) / 256, 256, 0, stream>>>(
        x, ws, out, ntok);
}